// GraphNeuralNetwork_83348135346322
// MI455X (gfx1250) — compile-verified
//
#include <hip/hip_runtime.h>
#include <math.h>

// ---------------------------------------------------------------------------
// GNN message-passing network, fused for MI455X (gfx1250, wave32, WMMA).
// All GEMMs run on V_WMMA_F32_16X16X4_F32 (f32 in/out, reference numerics).
// Edge MLP fully fused: gather x[src] (L2-resident) -> MLP -> atomic scatter.
// Weights are pre-swizzled in LDS into B-fragment order so each WMMA B operand
// is one aligned ds_load_b64; A rows are prefetched into registers ahead of
// the K-loop. An explicit compiler memory barrier at the top of each
// persistent tile loop stops LICM from hoisting the (loop-invariant) LDS
// weight fragments into registers, which previously caused scratch spills.
// ---------------------------------------------------------------------------

#define N_NODES 50000
#define N_EDGES 625000
#define D_NODE  64
#define D_EDGE  16
#define HDIM    128
#define NLAYERS 3
#define LN_EPS  1e-6f

// B-fragment-swizzled weight storage:
//   group g = k0/4 (4 K-rows), half h = lane>>4.
//   pair (W[4g+2h][c], W[4g+2h+1][c]) stored contiguously at
//   (g*2+h)*PSTRIDE + 2*c.  PSTRIDE=288 (pad from 256) => half-wave 1 is
//   shifted 32 banks from half-wave 0 -> conflict-free ds_load_b64.
#define PSTRIDE 288
#define SSTRIDE 132   // activation staging row stride (conflict-free)

typedef float v2f __attribute__((ext_vector_type(2)));
typedef float v8f __attribute__((ext_vector_type(8)));

static __device__ __forceinline__ v8f vzero8() {
  v8f z = {0.f, 0.f, 0.f, 0.f, 0.f, 0.f, 0.f, 0.f};
  return z;
}
static __device__ __forceinline__ v2f vzero2() {
  v2f z = {0.f, 0.f};
  return z;
}

// Stop LICM from hoisting LDS reads (weight fragments) across the persistent
// tile loop -- hoisting 100+ v2f fragments spills to scratch.
static __device__ __forceinline__ void no_hoist() {
  asm volatile("" ::: "memory");
}

// D = A(16x4 f32) * B(4x16 f32) + C(16x16 f32), one wave.
static __device__ __forceinline__ v8f wmma4(v2f a, v2f b, v8f c) {
  return __builtin_amdgcn_wmma_f32_16x16x4_f32(
      false, a, false, b, (short)0, c, false, false);
}

static __device__ __forceinline__ float gelu_tanh(float x) {
  float x3 = x * x * x;
  return 0.5f * x * (1.0f + tanhf(0.7978845608028654f * (x + 0.044715f * x3)));
}

// Scatter global weight W[K x 128] (row-major) into B-fragment order in LDS.
static __device__ __forceinline__ void load_w_swizzled(
    float* wl, const float* __restrict__ W, int K, int tid, int nthreads) {
  for (int i = tid; i < K * HDIM; i += nthreads) {
    int r = i >> 7, c = i & 127;
    wl[(((r >> 2) << 1) + ((r >> 1) & 1)) * PSTRIDE + (c << 1) + (r & 1)] = W[i];
  }
}

// ---------------------------------------------------------------------------
// x = gelu(node_features[N,64] @ W_in[64,128] + b_in)
// ---------------------------------------------------------------------------
__global__ void k_input_proj(const float* __restrict__ nf,
                             const float* __restrict__ W,
                             const float* __restrict__ b,
                             float* __restrict__ x) {
  extern __shared__ float lds[];
  float* wl = lds;  // 16 groups * 2 * PSTRIDE
  int tid = threadIdx.x;
  load_w_swizzled(wl, W, D_NODE, tid, blockDim.x);
  __syncthreads();

  int wave = tid >> 5, lane = tid & 31;
  int lo = lane & 15;
  int hi2 = (lane >> 4) << 1;   // K sub-offset of this half-wave
  int hi8 = (lane >> 4) << 3;   // D-row offset of this half-wave
  int hsel = lane >> 4;
  int wavesPerGrid = gridDim.x * (blockDim.x >> 5);
  int nTiles = N_NODES / 16;    // 3125 exactly

  for (int tile = blockIdx.x * (blockDim.x >> 5) + wave; tile < nTiles;
       tile += wavesPerGrid) {
    no_hoist();
    int rowBase = tile * 16;
    const float* arow = nf + (long)(rowBase + lo) * D_NODE;
    v2f areg[16];
#pragma unroll
    for (int g = 0; g < 16; ++g) areg[g] = *(const v2f*)(arow + 4 * g + hi2);

    v8f acc[8];
#pragma unroll
    for (int ct = 0; ct < 8; ++ct) acc[ct] = vzero8();
#pragma unroll
    for (int g = 0; g < 16; ++g) {
      v2f a = areg[g];
      const float* bp = wl + (g * 2 + hsel) * PSTRIDE + lo * 2;
#pragma unroll
      for (int ct = 0; ct < 8; ++ct)
        acc[ct] = wmma4(a, *(const v2f*)(bp + ct * 32), acc[ct]);
    }
#pragma unroll
    for (int ct = 0; ct < 8; ++ct) {
      int col = ct * 16 + lo;
      float bias = b[col];
#pragma unroll
      for (int v = 0; v < 8; ++v)
        x[(long)(rowBase + hi8 + v) * HDIM + col] = gelu_tanh(acc[ct][v] + bias);
    }
  }
}

// ---------------------------------------------------------------------------
// Fused edge MLP + scatter:
//   m = gelu(concat(x[src], ef) @ W1 + b1) @ W2 + b2 ; agg[dst] += m
// ---------------------------------------------------------------------------
__global__ void k_edge_mlp(const float* __restrict__ x,
                           const float* __restrict__ ef,
                           const int* __restrict__ eidx,  // [2*E]: src, dst
                           const float* __restrict__ W1,  // 144x128
                           const float* __restrict__ b1,
                           const float* __restrict__ W2,  // 128x128
                           const float* __restrict__ b2,
                           float* __restrict__ agg) {
  extern __shared__ float lds[];
  float* w1 = lds;                         // 36 groups * 2 * PSTRIDE
  float* w2 = w1 + 36 * 2 * PSTRIDE;       // 32 groups * 2 * PSTRIDE
  float* stageAll = w2 + 32 * 2 * PSTRIDE; // 8 waves * 16 * SSTRIDE
  int tid = threadIdx.x;
  load_w_swizzled(w1, W1, HDIM + D_EDGE, tid, blockDim.x);
  load_w_swizzled(w2, W2, HDIM, tid, blockDim.x);
  __syncthreads();

  int wave = tid >> 5, lane = tid & 31;
  int lo = lane & 15, hi2 = (lane >> 4) << 1, hi8 = (lane >> 4) << 3;
  int hsel = lane >> 4;
  float* stage = stageAll + wave * 16 * SSTRIDE;
  const int* src = eidx;
  const int* dst = eidx + N_EDGES;
  int nTiles = (N_EDGES + 15) / 16;  // 39063 (last tile half full)
  int wavesPerGrid = gridDim.x * (blockDim.x >> 5);

  for (int tile = blockIdx.x * (blockDim.x >> 5) + wave; tile < nTiles;
       tile += wavesPerGrid) {
    no_hoist();
    int eBase = tile * 16;
    int eMine = eBase + lo;                 // this lane's A-row edge
    bool ev = eMine < N_EDGES;
    int s = ev ? src[eMine] : 0;
    const float* xrow = x + (long)s * HDIM;
    const float* erow = ef + (long)eMine * D_EDGE;

    // Prefetch next tile's gathered row while this tile computes.
    int tileNext = tile + wavesPerGrid;
    if (tileNext < nTiles) {
      int eN = tileNext * 16 + lo;
      if (eN < N_EDGES) {
        const float* xn = x + (long)src[eN] * HDIM;
        __builtin_prefetch(xn, 0, 0);
        __builtin_prefetch(xn + 64, 0, 0);
      }
    }

    int drow[8];  // dst nodes of the rows this lane stores
#pragma unroll
    for (int v = 0; v < 8; ++v) {
      int e = eBase + hi8 + v;
      drow[v] = (e < N_EDGES) ? dst[e] : -1;
    }

    // A row (concat(x[src], ef)) -> registers, all loads issued up front.
    v2f areg[36];
#pragma unroll
    for (int g = 0; g < 32; ++g)
      areg[g] = ev ? *(const v2f*)(xrow + 4 * g + hi2) : vzero2();
#pragma unroll
    for (int g = 0; g < 4; ++g)
      areg[32 + g] = ev ? *(const v2f*)(erow + 4 * g + hi2) : vzero2();

    // ---- GEMM1: [16,144] x [144,128]
    v8f acc[8];
#pragma unroll
    for (int ct = 0; ct < 8; ++ct) acc[ct] = vzero8();
#pragma unroll
    for (int g = 0; g < 36; ++g) {
      v2f a = areg[g];
      const float* bp = w1 + (g * 2 + hsel) * PSTRIDE + lo * 2;
#pragma unroll
      for (int ct = 0; ct < 8; ++ct)
        acc[ct] = wmma4(a, *(const v2f*)(bp + ct * 32), acc[ct]);
    }
    // bias + gelu -> LDS stage (re-layout D -> A for GEMM2)
#pragma unroll
    for (int ct = 0; ct < 8; ++ct) {
      int col = ct * 16 + lo;
      float bias = b1[col];
#pragma unroll
      for (int v = 0; v < 8; ++v)
        stage[(hi8 + v) * SSTRIDE + col] = gelu_tanh(acc[ct][v] + bias);
    }

    // ---- GEMM2: [16,128] x [128,128]
    v8f acc2[8];
#pragma unroll
    for (int ct = 0; ct < 8; ++ct) acc2[ct] = vzero8();
#pragma unroll
    for (int g = 0; g < 32; ++g) {
      v2f a = *(const v2f*)(stage + lo * SSTRIDE + 4 * g + hi2);
      const float* bp = w2 + (g * 2 + hsel) * PSTRIDE + lo * 2;
#pragma unroll
      for (int ct = 0; ct < 8; ++ct)
        acc2[ct] = wmma4(a, *(const v2f*)(bp + ct * 32), acc2[ct]);
    }
    // bias + scatter-add into agg[dst]
#pragma unroll
    for (int ct = 0; ct < 8; ++ct) {
      int col = ct * 16 + lo;
      float bias = b2[col];
#pragma unroll
      for (int v = 0; v < 8; ++v) {
        if (drow[v] >= 0)
          atomicAdd(&agg[(long)drow[v] * HDIM + col], acc2[ct][v] + bias);
      }
    }
  }
}

// ---------------------------------------------------------------------------
// h = gelu(LayerNorm(concat(x, agg) @ Wu + bu)); x += h; agg = 0
// ---------------------------------------------------------------------------
__global__ void k_node_update(float* __restrict__ x,
                              float* __restrict__ agg,
                              const float* __restrict__ Wu,  // 256x128
                              const float* __restrict__ bu,
                              const float* __restrict__ lns,
                              const float* __restrict__ lnb) {
  extern __shared__ float lds[];
  float* wl = lds;                          // 64 groups * 2 * PSTRIDE
  float* stageAll = wl + 64 * 2 * PSTRIDE;  // 8 waves * 16 * SSTRIDE
  int tid = threadIdx.x;
  load_w_swizzled(wl, Wu, 2 * HDIM, tid, blockDim.x);
  __syncthreads();

  int wave = tid >> 5, lane = tid & 31;
  int lo = lane & 15, hi2 = (lane >> 4) << 1, hi8 = (lane >> 4) << 3;
  int hsel = lane >> 4;
  float* stage = stageAll + wave * 16 * SSTRIDE;
  int nTiles = N_NODES / 16;
  int wavesPerGrid = gridDim.x * (blockDim.x >> 5);

  for (int tile = blockIdx.x * (blockDim.x >> 5) + wave; tile < nTiles;
       tile += wavesPerGrid) {
    no_hoist();
    int rowBase = tile * 16;
    const float* xrow = x + (long)(rowBase + lo) * HDIM;
    const float* arow = agg + (long)(rowBase + lo) * HDIM;
    v8f acc[8];
#pragma unroll
    for (int ct = 0; ct < 8; ++ct) acc[ct] = vzero8();

    v2f areg[32];
    // K = 0..127 : x part
#pragma unroll
    for (int g = 0; g < 32; ++g) areg[g] = *(const v2f*)(xrow + 4 * g + hi2);
#pragma unroll
    for (int g = 0; g < 32; ++g) {
      v2f a = areg[g];
      const float* bp = wl + (g * 2 + hsel) * PSTRIDE + lo * 2;
#pragma unroll
      for (int ct = 0; ct < 8; ++ct)
        acc[ct] = wmma4(a, *(const v2f*)(bp + ct * 32), acc[ct]);
    }
    // K = 128..255 : agg part
#pragma unroll
    for (int g = 0; g < 32; ++g) areg[g] = *(const v2f*)(arow + 4 * g + hi2);
#pragma unroll
    for (int g = 0; g < 32; ++g) {
      v2f a = areg[g];
      const float* bp = wl + ((g + 32) * 2 + hsel) * PSTRIDE + lo * 2;
#pragma unroll
      for (int ct = 0; ct < 8; ++ct)
        acc[ct] = wmma4(a, *(const v2f*)(bp + ct * 32), acc[ct]);
    }

#pragma unroll
    for (int ct = 0; ct < 8; ++ct) {
      int col = ct * 16 + lo;
      float bias = bu[col];
#pragma unroll
      for (int v = 0; v < 8; ++v)
        stage[(hi8 + v) * SSTRIDE + col] = acc[ct][v] + bias;
    }
    if (lane < 16) {
      // LayerNorm + gelu + residual for row `lo`
      float mu = 0.f;
      for (int c = 0; c < HDIM; ++c) mu += stage[lo * SSTRIDE + c];
      mu *= (1.0f / HDIM);
      float var = 0.f;
      for (int c = 0; c < HDIM; ++c) {
        float d = stage[lo * SSTRIDE + c] - mu;
        var += d * d;
      }
      var *= (1.0f / HDIM);
      float inv = rsqrtf(var + LN_EPS);
      float* xo = x + (long)(rowBase + lo) * HDIM;
      for (int c = 0; c < HDIM; ++c) {
        float h = (stage[lo * SSTRIDE + c] - mu) * inv * lns[c] + lnb[c];
        xo[c] += gelu_tanh(h);
      }
    } else {
      // lanes 16-31: clear this tile's agg rows for the next layer
      float* ao = agg + (long)(rowBase + lo) * HDIM;
      for (int c = 0; c < HDIM; ++c) ao[c] = 0.f;
    }
  }
}

__global__ void k_zero(float* __restrict__ p, long n) {
  long i = (long)blockIdx.x * blockDim.x + threadIdx.x;
  long stride = (long)gridDim.x * blockDim.x;
  for (; i < n; i += stride) p[i] = 0.f;
}

__global__ void k_pool(const float* __restrict__ x, float* __restrict__ gsum) {
  int col = threadIdx.x;  // 128 threads/block, one column each
  float s = 0.f;
  for (int row = blockIdx.x; row < N_NODES; row += gridDim.x)
    s += x[(long)row * HDIM + col];
  atomicAdd(&gsum[col], s);
}

__global__ void k_readout(const float* __restrict__ gsum,
                          const float* __restrict__ W1, const float* __restrict__ b1,
                          const float* __restrict__ W2, const float* __restrict__ b2,
                          const float* __restrict__ W3, const float* __restrict__ b3,
                          float* __restrict__ out) {
  __shared__ float g[HDIM];
  __shared__ float t1[256];
  __shared__ float t2[HDIM];
  __shared__ float red[64];
  int tid = threadIdx.x;  // 256 threads
  if (tid < HDIM) g[tid] = gsum[tid];
  __syncthreads();
  {
    float s = b1[tid];
    for (int k = 0; k < HDIM; ++k) s += g[k] * W1[k * 256 + tid];
    t1[tid] = gelu_tanh(s);
  }
  __syncthreads();
  if (tid < HDIM) {
    float s = b2[tid];
    for (int k = 0; k < 256; ++k) s += t1[k] * W2[k * HDIM + tid];
    t2[tid] = gelu_tanh(s);
  }
  __syncthreads();
  if (tid < 64) red[tid] = t2[tid] * W3[tid] + t2[tid + 64] * W3[tid + 64];
  __syncthreads();
  if (tid == 0) {
    float s = 0.f;
    for (int k = 0; k < 64; ++k) s += red[k];
    out[0] = s + b3[0];
  }
}

extern "C" void kernel_launch(void* const* d_in, const int* in_sizes, int n_in,
                              void* d_out, int out_size, void* d_ws, size_t ws_size,
                              hipStream_t stream) {
  const float* nf   = (const float*)d_in[0];
  const float* ef   = (const float*)d_in[1];
  const int*   eidx = (const int*)d_in[2];
  const float* W_in = (const float*)d_in[3];
  const float* b_in = (const float*)d_in[4];
  const float* W_m1 = (const float*)d_in[5];
  const float* b_m1 = (const float*)d_in[6];
  const float* W_m2 = (const float*)d_in[7];
  const float* b_m2 = (const float*)d_in[8];
  const float* W_u  = (const float*)d_in[9];
  const float* b_u  = (const float*)d_in[10];
  const float* ln_s = (const float*)d_in[11];
  const float* ln_b = (const float*)d_in[12];
  const float* W_r1 = (const float*)d_in[13];
  const float* b_r1 = (const float*)d_in[14];
  const float* W_r2 = (const float*)d_in[15];
  const float* b_r2 = (const float*)d_in[16];
  const float* W_r3 = (const float*)d_in[17];
  const float* b_r3 = (const float*)d_in[18];
  float* out = (float*)d_out;

  // Workspace: x [N,H] + agg [N,H] + gsum [H]  (~51.3 MB)
  float* x    = (float*)d_ws;
  float* agg  = x + (size_t)N_NODES * HDIM;
  float* gsum = agg + (size_t)N_NODES * HDIM;

  const int ldsProj = (16 * 2 * PSTRIDE) * 4;
  const int ldsEdge = ((36 + 32) * 2 * PSTRIDE + 8 * 16 * SSTRIDE) * 4;
  const int ldsNode = (64 * 2 * PSTRIDE + 8 * 16 * SSTRIDE) * 4;
  hipFuncSetAttribute((const void*)k_input_proj,
                      hipFuncAttributeMaxDynamicSharedMemorySize, ldsProj);
  hipFuncSetAttribute((const void*)k_edge_mlp,
                      hipFuncAttributeMaxDynamicSharedMemorySize, ldsEdge);
  hipFuncSetAttribute((const void*)k_node_update,
                      hipFuncAttributeMaxDynamicSharedMemorySize, ldsNode);

  k_input_proj<<<391, 256, ldsProj, stream>>>(nf, W_in, b_in, x);
  k_zero<<<256, 256, 0, stream>>>(agg, (long)N_NODES * HDIM);
  k_zero<<<1, 128, 0, stream>>>(gsum, HDIM);
  for (int l = 0; l < NLAYERS; ++l) {
    k_edge_mlp<<<512, 256, ldsEdge, stream>>>(
        x, ef, eidx,
        W_m1 + (size_t)l * (HDIM + D_EDGE) * HDIM, b_m1 + l * HDIM,
        W_m2 + (size_t)l * HDIM * HDIM, b_m2 + l * HDIM, agg);
    k_node_update<<<256, 256, ldsNode, stream>>>(
        x, agg, W_u + (size_t)l * 2 * HDIM * HDIM, b_u + l * HDIM,
        ln_s + l * HDIM, ln_b + l * HDIM);
  }
  k_pool<<<200, 128, 0, stream>>>(x, gsum);
  k_readout<<<1, 256, 0, stream>>>(gsum, W_r1, b_r1, W_r2, b_r2, W_r3, b_r3, out);
}